// _CanonLayer_53343493816967
// MI455X (gfx1250) — compile-verified
//
#include <hip/hip_runtime.h>
#include <stdint.h>

// Problem constants (match reference: B=8, T=4096, C=2048, K=4, fp32, residual)
#define B_      8
#define T_      4096
#define C_      2048
#define K_      4
#define TCHUNK  64
#define THREADS 256
#define C4      (C_ / 4)          // 512 float4 lanes across C
#define CBLK    (C4 / THREADS)    // 2 blocks span one (b,t) row
#define NTCH    (T_ / TCHUNK)     // 64 time chunks

typedef float v4f __attribute__((ext_vector_type(4)));

__global__ __launch_bounds__(THREADS)
void canon_dconv_residual_kernel(const float* __restrict__ x,
                                 const float* __restrict__ w,
                                 float* __restrict__ out)
{
    // LDS staging for weights: per-thread 4 channels x 4 taps = 64B -> 16KB/block
    __shared__ float sw[THREADS * 16];

    const int tid  = threadIdx.x;
    const int cblk = blockIdx.x % CBLK;
    int       tmp  = blockIdx.x / CBLK;
    const int tch  = tmp % NTCH;
    const int b    = tmp / NTCH;

    const int c4 = cblk * THREADS + tid;   // float4 index along C
    const int c0 = c4 * 4;                 // first channel of this thread
    const int t0 = tch * TCHUNK;

    // ---- Stage weights via CDNA5 async global->LDS DMA path ----
    // Each lane copies 4 x 16B (the 4-tap rows of its 4 channels).
    {
        const float* wg   = w + (size_t)c0 * K_;                    // 64B contiguous
        unsigned     lds0 = (unsigned)(uintptr_t)(&sw[tid * 16]);   // LDS byte addr
        #pragma unroll
        for (int j = 0; j < 4; ++j) {
            unsigned     ldsa = lds0 + (unsigned)(j * 16);
            const float* ga   = wg + j * 4;
            asm volatile("global_load_async_to_lds_b128 %0, %1, off"
                         :: "v"(ldsa), "v"(ga) : "memory");
        }
        asm volatile("s_wait_asynccnt 0" ::: "memory");  // our LDS data is ready
    }

    // Read back per-channel tap rows from LDS (ds_load_b128) and transpose
    // into per-tap vectors across the 4 channels. Fold residual into tap 3.
    const float* swt = &sw[tid * 16];
    v4f ch0 = *(const v4f*)(swt + 0);
    v4f ch1 = *(const v4f*)(swt + 4);
    v4f ch2 = *(const v4f*)(swt + 8);
    v4f ch3 = *(const v4f*)(swt + 12);
    v4f w0  = (v4f){ch0.x, ch1.x, ch2.x, ch3.x};
    v4f w1  = (v4f){ch0.y, ch1.y, ch2.y, ch3.y};
    v4f w2  = (v4f){ch0.z, ch1.z, ch2.z, ch3.z};
    v4f w3p = (v4f){ch0.w + 1.0f, ch1.w + 1.0f, ch2.w + 1.0f, ch3.w + 1.0f};

    const size_t base = ((size_t)b * T_ + t0) * C_ + c0;
    const float* xrow = x + base;
    float*       orow = out + base;

    // ---- Causal history (register sliding window) ----
    v4f xm3, xm2, xm1;
    if (t0 == 0) {
        xm3 = (v4f){0.f, 0.f, 0.f, 0.f};
        xm2 = xm3;
        xm1 = xm3;
    } else {
        xm3 = __builtin_nontemporal_load((const v4f*)(xrow - 3 * C_));
        xm2 = __builtin_nontemporal_load((const v4f*)(xrow - 2 * C_));
        xm1 = __builtin_nontemporal_load((const v4f*)(xrow - 1 * C_));
    }

    // ---- Stream: each element loaded exactly once, stored exactly once.
    // Non-temporal: 537MB single-pass stream, no L2 reuse to protect.
    #pragma unroll 4
    for (int i = 0; i < TCHUNK; ++i) {
        // lookahead prefetch 8 rows (64KB) down our stream; speculative,
        // OOB translations are silently dropped per ISA prefetch semantics
        __builtin_prefetch(xrow + 8 * C_, 0, 0);

        v4f cur = __builtin_nontemporal_load((const v4f*)xrow);
        v4f o   = w0 * xm3 + w1 * xm2 + w2 * xm1 + w3p * cur; // residual folded
        __builtin_nontemporal_store(o, (v4f*)orow);

        xm3 = xm2;
        xm2 = xm1;
        xm1 = cur;
        xrow += C_;
        orow += C_;
    }
}

extern "C" void kernel_launch(void* const* d_in, const int* in_sizes, int n_in,
                              void* d_out, int out_size, void* d_ws, size_t ws_size,
                              hipStream_t stream) {
    (void)in_sizes; (void)n_in; (void)d_ws; (void)ws_size; (void)out_size;
    const float* x  = (const float*)d_in[0];   // (B,T,C) fp32
    const float* w  = (const float*)d_in[1];   // (C,1,K) fp32
    float*       o  = (float*)d_out;           // (B,T,C) fp32

    const int grid = B_ * NTCH * CBLK;         // 8*64*2 = 1024 blocks
    canon_dconv_residual_kernel<<<grid, THREADS, 0, stream>>>(x, w, o);
}